// ARX_27049704030515
// MI455X (gfx1250) — compile-verified
//
#include <hip/hip_runtime.h>

// ARX (16,16) over N = 2^20 steps, parallelized as a hierarchical state-space
// scan. All heavy phases are f32 16x16 GEMMs on V_WMMA_F32_16X16X4_F32.
//
// Hierarchy: steps(2^20) -> blocks(65536, 16 steps) -> groups(4096)
//            -> supergroups(256) -> serial 256-step micro-scan.
//
// d_ws usage: ~9 MB, fully (re)written every call.

typedef __attribute__((ext_vector_type(2))) float v2f;
typedef __attribute__((ext_vector_type(8))) float v8f;

#define NSTEPS (1 << 20)
#define NBLK (NSTEPS / 16) /* 65536 */
#define NGRP (NBLK / 16)   /* 4096  */
#define NSUP (NGRP / 16)   /* 256   */

// constant-region offsets (floats) inside d_ws
#define OFF_P 0
#define OFF_Q 256
#define OFF_G 512
#define OFF_POWS0 768                       /* M^0..M^16 (17 mats) */
#define OFF_POWS1 (OFF_POWS0 + 17 * 256)    /* T1^0..T1^16, T1=M^16 */
#define CONST_FLOATS 16384

#define OFF_F (CONST_FLOATS)
#define OFF_V0 (OFF_F + NSTEPS)
#define OFF_V1 (OFF_V0 + NSTEPS)
#define OFF_V2 (OFF_V1 + NGRP * 16)
#define OFF_S2 (OFF_V2 + NSUP * 16)
#define OFF_S1 (OFF_S2 + NSUP * 16)

// ---------------- WMMA helpers (wave32, f32 16x16x4) ----------------
// A fragment: 16x4 (MxK), operand stored row-major with ld=16.
// ISA layout: lanes 0-15 hold (M=lane, K=k0,k0+1); lanes 16-31 (M=lane-16, K=k0+2,k0+3).
__device__ __forceinline__ v2f load_a_rm16(const float* A, int k0) {
  int lane = threadIdx.x & 31;
  int m = lane & 15;
  int kk = k0 + ((lane >> 4) << 1);
  v2f a;
  a.x = A[m * 16 + kk];
  a.y = A[m * 16 + kk + 1];
  return a;
}

// B fragment: 4x16 (KxN), operand column-major: element (k,n) at B[n*ldb+k].
__device__ __forceinline__ v2f load_b_cm(const float* B, int ldb, int k0) {
  int lane = threadIdx.x & 31;
  int n = lane & 15;
  int kk = k0 + ((lane >> 4) << 1);
  v2f b;
  b.x = B[n * ldb + kk];
  b.y = B[n * ldb + kk + 1];
  return b;
}

// C/D 16x16 f32: VGPR r: lanes 0-15 -> M=r, lanes 16-31 -> M=r+8 (ISA 7.12.2).
__device__ __forceinline__ void store_c_cm(float* C, int ldc, v8f c) {
  int lane = threadIdx.x & 31;
  int n = lane & 15;
  int mbase = (lane >> 4) << 3;
#pragma unroll
  for (int r = 0; r < 8; ++r) C[n * ldc + mbase + r] = c[r];
}

// C(16x16) += A_rowmajor(16x16) * B_colmajor(16x16 slice, ld=ldb)
__device__ __forceinline__ v8f mm16(const float* A, const float* B, int ldb, v8f c) {
#pragma unroll
  for (int k0 = 0; k0 < 16; k0 += 4) {
    v2f a = load_a_rm16(A, k0);
    v2f b = load_b_cm(B, ldb, k0);
    c = __builtin_amdgcn_wmma_f32_16x16x4_f32(false, a, false, b, (short)0, c,
                                              false, false);
  }
  return c;
}

// ---------------- K0: build operator matrices on device ----------------
__global__ void __launch_bounds__(256) k_setup(const float* A_w, float* cst) {
  __shared__ float a[16];
  __shared__ float h[16];
  __shared__ float Pm[256];
  __shared__ float Mm[256];
  __shared__ float tmp[256];
  int t = threadIdx.x;
  if (t < 16) a[t] = A_w[t];
  __syncthreads();
  if (t == 0) {
    h[0] = 1.0f; // impulse response of the AR part
    for (int m = 1; m < 16; ++m) {
      float acc = 0.0f;
      for (int i = 0; i < 16; ++i) {
        int mm = m - 1 - i;
        if (mm >= 0) acc += a[i] * h[mm];
      }
      h[m] = acc;
    }
  }
  __syncthreads();
  if (t < 16) {
    // column t of P: simulate 16 homogeneous steps from s = e_t
    float s[16];
    for (int i = 0; i < 16; ++i) s[i] = (i == t) ? 1.0f : 0.0f;
    for (int step = 0; step < 16; ++step) {
      float y = 0.0f;
      for (int i = 0; i < 16; ++i) y += a[i] * s[i];
      Pm[step * 16 + t] = y;
      for (int i = 15; i > 0; --i) s[i] = s[i - 1];
      s[0] = y;
    }
  }
  __syncthreads();
  {
    int r = t >> 4, cc = t & 15;
    cst[OFF_P + t] = Pm[t];
    cst[OFF_Q + t] = (r >= cc) ? h[r - cc] : 0.0f; // Q[t][r]=h[t-r]
    int hg = 15 - r - cc;
    cst[OFF_G + t] = (hg >= 0) ? h[hg] : 0.0f;     // G[row][r]=h[15-row-r]
    Mm[t] = Pm[(15 - r) * 16 + cc];                // M = revrows(P) = C^16
    cst[OFF_POWS0 + t] = (r == cc) ? 1.0f : 0.0f;  // M^0 = I
    cst[OFF_POWS0 + 256 + t] = Mm[t];
    tmp[t] = Mm[t];
  }
  __syncthreads();
  for (int p = 2; p <= 16; ++p) { // Pows0[p] = Pows0[p-1] * M
    int r = t >> 4, cc = t & 15;
    float acc = 0.0f;
    for (int i = 0; i < 16; ++i) acc += tmp[r * 16 + i] * Mm[i * 16 + cc];
    __syncthreads();
    tmp[t] = acc;
    cst[OFF_POWS0 + p * 256 + t] = acc;
    __syncthreads();
  }
  {
    int r = t >> 4, cc = t & 15;
    Mm[t] = tmp[t]; // T1 = M^16
    cst[OFF_POWS1 + t] = (r == cc) ? 1.0f : 0.0f;
    cst[OFF_POWS1 + 256 + t] = tmp[t];
  }
  __syncthreads();
  for (int p = 2; p <= 16; ++p) { // Pows1[p] = Pows1[p-1] * T1
    int r = t >> 4, cc = t & 15;
    float acc = 0.0f;
    for (int i = 0; i < 16; ++i) acc += tmp[r * 16 + i] * Mm[i * 16 + cc];
    __syncthreads();
    tmp[t] = acc;
    cst[OFF_POWS1 + p * 256 + t] = acc;
    __syncthreads();
  }
}

// ---------------- K1: FIR feedforward f[k] and per-block drive v0 = G f ----
__global__ void __launch_bounds__(256) k_fir(const float* __restrict__ u,
                                             const float* __restrict__ B_w,
                                             const float* __restrict__ G,
                                             float* __restrict__ F,
                                             float* __restrict__ V0) {
  __shared__ float su[272];
  __shared__ float sf[256];
  __shared__ float sb[16];
  int t = threadIdx.x;
  int k0 = blockIdx.x * 256;
  if (t < 16) sb[t] = B_w[t];
  su[t] = u[k0 + t];
  if (t < 16) su[256 + t] = u[k0 + 256 + t];
  __syncthreads();
  float acc = 0.0f;
#pragma unroll
  for (int i = 0; i < 16; ++i) acc += sb[i] * su[t + 15 - i]; // f[k]=sum b_i u[k+15-i]
  sf[t] = acc;
  F[k0 + t] = acc;
  __syncthreads();
  int jl = t >> 4, r = t & 15;
  float v = 0.0f;
#pragma unroll
  for (int tt = 0; tt < 16; ++tt) v += G[r * 16 + tt] * sf[jl * 16 + tt];
  V0[k0 + t] = v; // V0[16*j + r], blocks contiguous
}

// ---------------- K2/K3: aggregate 16 child drives -> parent drive ---------
// Vout(16 x nGroups) = [T^15 | T^14 | ... | T^0] x Vin_stacked(256 x nGroups)
__global__ void __launch_bounds__(256) k_agg(const float* __restrict__ Pows,
                                             const float* __restrict__ Vin,
                                             float* __restrict__ Vout) {
  int wave = threadIdx.x >> 5;
  int g0 = (blockIdx.x * 8 + wave) * 16; // 16 groups (columns) per wave
  const float* Bbase = Vin + (size_t)g0 * 256;
  v8f c = {};
  for (int rb = 0; rb < 16; ++rb)
    c = mm16(Pows + (15 - rb) * 256, Bbase + rb * 16, 256, c);
  store_c_cm(Vout + g0 * 16, 16, c);
}

// ---------------- K4: tiny serial scan over 256 supergroups ----------------
__global__ void k_topscan(const float* __restrict__ T2,
                          const float* __restrict__ V2,
                          float* __restrict__ S2) {
  __shared__ float s[16];
  int t = threadIdx.x;
  if (t < 16) s[t] = 0.0f;
  __syncthreads();
  for (int c = 0; c < NSUP; ++c) {
    if (t < 16) S2[c * 16 + t] = s[t]; // state at supergroup start
    __syncthreads();
    float sn = 0.0f;
    if (t < 16) {
      for (int i = 0; i < 16; ++i) sn += T2[t * 16 + i] * s[i];
      sn += V2[c * 16 + t];
    }
    __syncthreads();
    if (t < 16) s[t] = sn;
    __syncthreads();
  }
}

// ---------------- K5: expand parent-start states -> 16 child-start states --
// For parent tile of 16: X(256x16) = PS * Sigma + TriL * Vstacked, where
// PS block t = T^t, TriL block (t,r) = T^{t-1-r} (strict block-lower).
__global__ void __launch_bounds__(256) k_expand(const float* __restrict__ Pows,
                                                const float* __restrict__ Sin,
                                                const float* __restrict__ Vin,
                                                float* __restrict__ Sout) {
  int wave = threadIdx.x >> 5;
  int p0 = blockIdx.x * 16;
  const float* Bsig = Sin + (size_t)p0 * 16;  // 16x16 col-major ld16
  const float* Bv = Vin + (size_t)p0 * 256;   // 256x16 col-major ld256
#pragma unroll 1
  for (int mi = 0; mi < 2; ++mi) {
    int mt = wave + mi * 8; // row-block; interleaved for triangular balance
    v8f c = {};
    c = mm16(Pows + mt * 256, Bsig, 16, c); // T^mt * sigma (T^0 = I)
    for (int rb = 0; rb < mt; ++rb)
      c = mm16(Pows + (mt - 1 - rb) * 256, Bv + rb * 16, 256, c);
    store_c_cm(Sout + (size_t)p0 * 256 + mt * 16, 256, c);
  }
}

// ---------------- K6: expand group states -> block states -> outputs -------
__global__ void __launch_bounds__(256) k_final(
    const float* __restrict__ Pows0, const float* __restrict__ Pmat,
    const float* __restrict__ Qmat, const float* __restrict__ S1,
    const float* __restrict__ V0, const float* __restrict__ F,
    float* __restrict__ Y) {
  __shared__ float X[4096]; // 16 KB: block-start states for 256 blocks
  int wave = threadIdx.x >> 5;
  int p0 = blockIdx.x * 16; // 16 groups => 256 blocks => 4096 steps
  const float* Bsig = S1 + (size_t)p0 * 16;
  const float* Bv = V0 + (size_t)p0 * 256;
#pragma unroll 1
  for (int mi = 0; mi < 2; ++mi) {
    int mt = wave + mi * 8;
    v8f c = {};
    c = mm16(Pows0 + mt * 256, Bsig, 16, c);
    for (int rb = 0; rb < mt; ++rb)
      c = mm16(Pows0 + (mt - 1 - rb) * 256, Bv + rb * 16, 256, c);
    store_c_cm(&X[mt * 16], 256, c); // X col-major per parent
  }
  __syncthreads();
  // Stage 2: Y(16 x 256 blocks) = P * S0tile + Q * Ftile
  const float* Fbase = F + (size_t)p0 * 256;
  float* Ybase = Y + (size_t)p0 * 256;
#pragma unroll 1
  for (int ci = 0; ci < 2; ++ci) {
    int ct = wave * 2 + ci; // col-tile of 16 blocks
    v8f c = {};
    c = mm16(Pmat, &X[ct * 256], 16, c);       // P * block-start states (LDS)
    c = mm16(Qmat, Fbase + ct * 256, 16, c);   // Q * f-block
    store_c_cm(Ybase + ct * 256, 16, c);       // y contiguous in time
  }
}

extern "C" void kernel_launch(void* const* d_in, const int* in_sizes, int n_in,
                              void* d_out, int out_size, void* d_ws,
                              size_t ws_size, hipStream_t stream) {
  const float* u = (const float*)d_in[0];
  const float* A_w = (const float*)d_in[1];
  const float* B_w = (const float*)d_in[2];
  float* ws = (float*)d_ws;
  float* Y = (float*)d_out;

  float* cst = ws;
  float* F = ws + OFF_F;
  float* V0 = ws + OFF_V0;
  float* V1 = ws + OFF_V1;
  float* V2 = ws + OFF_V2;
  float* S2 = ws + OFF_S2;
  float* S1 = ws + OFF_S1;

  k_setup<<<1, 256, 0, stream>>>(A_w, cst);
  k_fir<<<NSTEPS / 256, 256, 0, stream>>>(u, B_w, cst + OFF_G, F, V0);
  // blocks -> groups (4096 cols / 128 per WG = 32 WGs), uses M powers
  k_agg<<<NGRP / 128, 256, 0, stream>>>(cst + OFF_POWS0, V0, V1);
  // groups -> supergroups (256 cols / 128 = 2 WGs), uses T1 powers
  k_agg<<<NSUP / 128, 256, 0, stream>>>(cst + OFF_POWS1, V1, V2);
  // serial 256-step scan with T2 = T1^16
  k_topscan<<<1, 32, 0, stream>>>(cst + OFF_POWS1 + 16 * 256, V2, S2);
  // supergroup states -> group states (256 parents / 16 = 16 WGs)
  k_expand<<<NSUP / 16, 256, 0, stream>>>(cst + OFF_POWS1, S2, V1, S1);
  // group states -> block states -> outputs (4096 parents / 16 = 256 WGs)
  k_final<<<NGRP / 16, 256, 0, stream>>>(cst + OFF_POWS0, cst + OFF_P,
                                         cst + OFF_Q, S1, V0, F, Y);
}